// DWFGCN_39908836114942
// MI455X (gfx1250) — compile-verified
//
#include <hip/hip_runtime.h>
#include <hip/hip_bf16.h>

// ---------------------------------------------------------------------------
// MI455X (gfx1250, wave32) implementation of the DWFGCN reference.
// Dense GEMMs use V_WMMA_F32_16X16X4_F32 (exact fp32) -- the workload is
// L2/atomic bound (h-matrices fit in the 192MB L2), so fp32 matrix math is
// free and keeps numerics identical to the reference.
// ---------------------------------------------------------------------------

typedef float v2f __attribute__((ext_vector_type(2)));
typedef float v8f __attribute__((ext_vector_type(8)));

// 8-arg form: (neg_a, A, neg_b, B, c_mod, C, reuse_a, reuse_b)
#define WMMA_F32(a, b, c) \
  __builtin_amdgcn_wmma_f32_16x16x4_f32(false, (a), false, (b), (short)0, (c), false, false)

// ---------------------------------------------------------------------------
// Small utility kernels
// ---------------------------------------------------------------------------
__global__ void fill_f32(float* __restrict__ p, float v, int n) {
  int i = blockIdx.x * blockDim.x + threadIdx.x;
  if (i < n) p[i] = v;
}

__global__ void rsqrt_f32(float* __restrict__ p, int n) {
  int i = blockIdx.x * blockDim.x + threadIdx.x;
  if (i < n) {
    float d = p[i];
    p[i] = (d > 0.f) ? rsqrtf(fmaxf(d, 1e-12f)) : 0.f;
  }
}

// deg[dst[e]] += w[e]  (w == nullptr -> 1.0f). Self-loop handled by init=1.0.
__global__ void degree_acc(const int* __restrict__ dst, const float* __restrict__ w,
                           float* __restrict__ dv, int E) {
  for (int e = blockIdx.x * blockDim.x + threadIdx.x; e < E;
       e += gridDim.x * blockDim.x)
    atomicAdd(&dv[dst[e]], w ? w[e] : 1.0f);
}

// Precompute TSK antecedent coefficient matrices, padded to 32 rules:
//   P[r][d] = c/(v*v)   (coeff of x)
//   Q[r][d] = -0.5/(v*v) (coeff of x^2)
//   kc[r]   = sum_d c^2/(v*v)
//   Cp      = c padded with zero rows (for the defuzz GEMM)
__global__ void prep_fuzz(const float* __restrict__ c, const float* __restrict__ v,
                          int R, int D, float* __restrict__ P, float* __restrict__ Q,
                          float* __restrict__ kc, float* __restrict__ Cp) {
  int r = threadIdx.x;
  if (r >= 32) return;
  if (r < R) {
    float acc = 0.f;
    for (int d = 0; d < D; ++d) {
      float vv = v[r * D + d];
      float iv = 1.0f / (vv * vv);
      float cc = c[r * D + d];
      P[r * D + d] = cc * iv;
      Q[r * D + d] = -0.5f * iv;
      Cp[r * D + d] = cc;
      acc += cc * cc * iv;
    }
    kc[r] = acc;
  } else {
    for (int d = 0; d < D; ++d) {
      P[r * D + d] = 0.f; Q[r * D + d] = 0.f; Cp[r * D + d] = 0.f;
    }
    kc[r] = 0.f;
  }
}

// ---------------------------------------------------------------------------
// fp32 WMMA GEMM: C[M x Nc] = A[M x K] @ B[K x Nc], row-major.
// One wave computes a 16x64 strip (4 accumulators). M % 16 == 0 (N=100000),
// Nc % 64 == 0, K % 4 == 0.
// A frag (16x4): lane<16 -> K=k+{0,1}; lane>=16 -> K=k+{2,3}; M = lane&15.
// B frag (4x16): vgpr j: lane<16 -> row k+j, lane>=16 -> row k+j+2; N = lane&15.
// C/D (16x16):   vgpr v: lane<16 -> M=v, lane>=16 -> M=v+8; N = lane&15.
// ---------------------------------------------------------------------------
__global__ __launch_bounds__(32)
void gemm_wmma_f32(const float* __restrict__ A, int lda,
                   const float* __restrict__ B, int ldb,
                   float* __restrict__ C, int ldc, int K) {
  const int lane = threadIdx.x & 31;
  const int row0 = blockIdx.x << 4;
  const int col0 = blockIdx.y << 6;
  const int m  = row0 + (lane & 15);
  const int kh = (lane >> 4) << 1;   // 0 or 2
  const int nl = lane & 15;

  v8f acc0 = {}, acc1 = {}, acc2 = {}, acc3 = {};
  for (int k = 0; k < K; k += 4) {
    const int k0 = k + kh;
    v2f a;
    a.x = A[(size_t)m * lda + k0];
    a.y = A[(size_t)m * lda + k0 + 1];
    const float* Br0 = B + (size_t)k0 * ldb + col0 + nl;
    const float* Br1 = Br0 + ldb;
    v2f b0, b1, b2, b3;
    b0.x = Br0[0];  b0.y = Br1[0];
    b1.x = Br0[16]; b1.y = Br1[16];
    b2.x = Br0[32]; b2.y = Br1[32];
    b3.x = Br0[48]; b3.y = Br1[48];
    acc0 = WMMA_F32(a, b0, acc0);
    acc1 = WMMA_F32(a, b1, acc1);
    acc2 = WMMA_F32(a, b2, acc2);
    acc3 = WMMA_F32(a, b3, acc3);
  }
  const int srow = row0 + ((lane >> 4) << 3);
#pragma unroll
  for (int vv = 0; vv < 8; ++vv) {
    float* crow = C + (size_t)(srow + vv) * ldc + col0 + nl;
    crow[0]  = acc0[vv];
    crow[16] = acc1[vv];
    crow[32] = acc2[vv];
    crow[48] = acc3[vv];
  }
}

// ---------------------------------------------------------------------------
// Fused fuzzify "z" kernel:  Z[n][r] = x_n . P_r + (x_n*x_n) . Q_r - 0.5*kc[r]
// One wave -> 16 nodes x 32 rules (2 accumulators); 4 WMMAs per K-step.
// Elementwise square of the A fragment is layout-preserving, so both GEMMs
// accumulate into the SAME WMMA accumulator.
// ---------------------------------------------------------------------------
__global__ __launch_bounds__(32)
void fuzz_z_wmma(const float* __restrict__ X, int K,
                 const float* __restrict__ P, const float* __restrict__ Q,
                 const float* __restrict__ kc, float* __restrict__ Z, int R) {
  const int lane = threadIdx.x & 31;
  const int row0 = blockIdx.x << 4;
  const int m  = row0 + (lane & 15);
  const int kh = (lane >> 4) << 1;
  const int nl = lane & 15;

  v8f acc0 = {}, acc1 = {};
  for (int k = 0; k < K; k += 4) {
    const int k0 = k + kh;
    v2f a, ax;
    a.x = X[(size_t)m * K + k0];
    a.y = X[(size_t)m * K + k0 + 1];
    ax.x = a.x * a.x;
    ax.y = a.y * a.y;
    v2f p0, q0, p1, q1;
    p0.x = P[(size_t)nl * K + k0];        p0.y = P[(size_t)nl * K + k0 + 1];
    q0.x = Q[(size_t)nl * K + k0];        q0.y = Q[(size_t)nl * K + k0 + 1];
    p1.x = P[(size_t)(nl + 16) * K + k0]; p1.y = P[(size_t)(nl + 16) * K + k0 + 1];
    q1.x = Q[(size_t)(nl + 16) * K + k0]; q1.y = Q[(size_t)(nl + 16) * K + k0 + 1];
    acc0 = WMMA_F32(a,  p0, acc0);
    acc0 = WMMA_F32(ax, q0, acc0);
    acc1 = WMMA_F32(a,  p1, acc1);
    acc1 = WMMA_F32(ax, q1, acc1);
  }
  const int srow = row0 + ((lane >> 4) << 3);
#pragma unroll
  for (int vv = 0; vv < 8; ++vv) {
    float* zrow = Z + (size_t)(srow + vv) * 32;
    const int r0 = nl, r1 = nl + 16;
    zrow[r0] = (r0 < R) ? (acc0[vv] - 0.5f * kc[r0]) : -1e30f;
    zrow[r1] = (r1 < R) ? (acc1[vv] - 0.5f * kc[r1]) : -1e30f;
  }
}

// Row softmax over 32 (padded) rules: one wave32 per node.
__global__ void softmax_rows32(float* __restrict__ Z, int N) {
  int wid = (blockIdx.x * blockDim.x + threadIdx.x) >> 5;
  int lane = threadIdx.x & 31;
  if (wid >= N) return;
  float* row = Z + (size_t)wid * 32;
  float v = row[lane];
  float m = v;
#pragma unroll
  for (int o = 16; o > 0; o >>= 1) m = fmaxf(m, __shfl_xor(m, o, 32));
  float e = __expf(v - m);          // padded lanes: exp(-1e30 - m) -> 0
  float s = e;
#pragma unroll
  for (int o = 16; o > 0; o >>= 1) s += __shfl_xor(s, o, 32);
  row[lane] = e / s;
}

// ---------------------------------------------------------------------------
// Edge scatter: OUT[dst] += H[src] * (dinv[src]*w*dinv[dst]).
// F/4 lanes per edge, float4 gather (h fits in L2) + global_atomic_add_f32.
// ---------------------------------------------------------------------------
__global__ void scatter_edges(const int* __restrict__ ei, const float* __restrict__ w,
                              const float* __restrict__ dinv, const float* __restrict__ H,
                              float* __restrict__ OUT, int E, int fshift) {
  const int tpeShift = fshift - 2;             // threads per edge = F/4
  const int tpeMask = (1 << tpeShift) - 1;
  const long long total = (long long)E << tpeShift;
  const long long stride = (long long)gridDim.x * blockDim.x;
  for (long long t = blockIdx.x * (long long)blockDim.x + threadIdx.x; t < total;
       t += stride) {
    const int e  = (int)(t >> tpeShift);
    const int fo = (int)(t & tpeMask) << 2;
    const int s = ei[e];
    const int d = ei[E + e];
    const float ww = w ? w[e] : 1.0f;
    const float norm = dinv[s] * ww * dinv[d];
    const float4 hv = *(const float4*)(H + ((size_t)s << fshift) + fo);
    float* o = OUT + ((size_t)d << fshift) + fo;
    atomicAdd(o + 0, hv.x * norm);
    atomicAdd(o + 1, hv.y * norm);
    atomicAdd(o + 2, hv.z * norm);
    atomicAdd(o + 3, hv.w * norm);
  }
}

// Self-loop contribution (+ bias): OUT += H * dinv^2 + b. Runs after the
// atomic scatter kernel has drained, so plain RMW is fine.
__global__ void selfloop_bias(const float* __restrict__ H, const float* __restrict__ dinv,
                              const float* __restrict__ bias, float* __restrict__ OUT,
                              int N, int fshift) {
  const int F = 1 << fshift;
  const long long total = (long long)N << fshift;
  const long long stride = (long long)gridDim.x * blockDim.x;
  for (long long i = blockIdx.x * (long long)blockDim.x + threadIdx.x; i < total;
       i += stride) {
    const int n = (int)(i >> fshift);
    const int f = (int)(i & (F - 1));
    const float dvv = dinv[n];
    OUT[i] += H[i] * dvv * dvv + bias[f];
  }
}

// ---------------------------------------------------------------------------
extern "C" void kernel_launch(void* const* d_in, const int* in_sizes, int n_in,
                              void* d_out, int out_size, void* d_ws, size_t ws_size,
                              hipStream_t stream) {
  const float* x  = (const float*)d_in[0];
  const int*   ei = (const int*)d_in[1];
  const float* ew = (const float*)d_in[2];
  const float* W1 = (const float*)d_in[3];
  const float* b1 = (const float*)d_in[4];
  const float* W2 = (const float*)d_in[5];
  const float* b2 = (const float*)d_in[6];
  const float* c1 = (const float*)d_in[7];
  const float* v1 = (const float*)d_in[8];
  const float* c2 = (const float*)d_in[9];
  const float* v2 = (const float*)d_in[10];
  float* out = (float*)d_out;

  const int DIN = 128, H1 = 128, H2 = 64, R = 30;
  const int N = in_sizes[0] / DIN;   // 100000 (multiple of 16)
  const int E = in_sizes[2];         // 1600000

  // Workspace layout (floats)
  float* ws = (float*)d_ws;
  size_t off = 0;
  float* bufA = ws + off; off += (size_t)N * H1;   // h1 / defuzz1
  float* bufB = ws + off; off += (size_t)N * H1;   // agg1 / h2
  float* bufC = ws + off; off += (size_t)N * 32;   // z / firing strengths
  float* dv   = ws + off; off += (size_t)N;        // deg -> dinv (reused)
  float* P1  = ws + off; off += 32 * H1;
  float* Q1  = ws + off; off += 32 * H1;
  float* C1p = ws + off; off += 32 * H1;
  float* kc1 = ws + off; off += 32;
  float* P2  = ws + off; off += 32 * H2;
  float* Q2  = ws + off; off += 32 * H2;
  float* C2p = ws + off; off += 32 * H2;
  float* kc2 = ws + off; off += 32;
  (void)ws_size; (void)n_in; (void)out_size;

  prep_fuzz<<<1, 32, 0, stream>>>(c1, v1, R, H1, P1, Q1, kc1, C1p);
  prep_fuzz<<<1, 32, 0, stream>>>(c2, v2, R, H2, P2, Q2, kc2, C2p);

  // ---------------- layer 1: gc1 (with edge weights) ----------------
  fill_f32<<<(N + 255) / 256, 256, 0, stream>>>(dv, 1.0f, N);          // self-loop deg
  degree_acc<<<1024, 256, 0, stream>>>(ei + E, ew, dv, E);
  rsqrt_f32<<<(N + 255) / 256, 256, 0, stream>>>(dv, N);
  gemm_wmma_f32<<<dim3(N / 16, H1 / 64), 32, 0, stream>>>(x, DIN, W1, H1, bufA, H1, DIN);
  hipMemsetAsync(bufB, 0, (size_t)N * H1 * sizeof(float), stream);
  scatter_edges<<<4096, 256, 0, stream>>>(ei, ew, dv, bufA, bufB, E, 7);
  selfloop_bias<<<4096, 256, 0, stream>>>(bufA, dv, b1, bufB, N, 7);

  // ---------------- fuzzify 1 ----------------
  fuzz_z_wmma<<<N / 16, 32, 0, stream>>>(bufB, H1, P1, Q1, kc1, bufC, R);
  softmax_rows32<<<(N * 32 + 255) / 256, 256, 0, stream>>>(bufC, N);
  gemm_wmma_f32<<<dim3(N / 16, H1 / 64), 32, 0, stream>>>(bufC, 32, C1p, H1, bufA, H1, 32);

  // ---------------- layer 2: gc2 (unit edge weights) ----------------
  gemm_wmma_f32<<<dim3(N / 16, H2 / 64), 32, 0, stream>>>(bufA, H1, W2, H2, bufB, H2, H1);
  fill_f32<<<(N + 255) / 256, 256, 0, stream>>>(dv, 1.0f, N);
  degree_acc<<<1024, 256, 0, stream>>>(ei + E, (const float*)nullptr, dv, E);
  rsqrt_f32<<<(N + 255) / 256, 256, 0, stream>>>(dv, N);
  hipMemsetAsync(out, 0, (size_t)N * H2 * sizeof(float), stream);
  scatter_edges<<<4096, 256, 0, stream>>>(ei, (const float*)nullptr, dv, bufB, out, E, 6);
  selfloop_bias<<<4096, 256, 0, stream>>>(bufB, dv, b2, out, N, 6);

  // ---------------- fuzzify 2 (writes final output) ----------------
  fuzz_z_wmma<<<N / 16, 32, 0, stream>>>(out, H2, P2, Q2, kc2, bufC, R);
  softmax_rows32<<<(N * 32 + 255) / 256, 256, 0, stream>>>(bufC, N);
  gemm_wmma_f32<<<dim3(N / 16, H2 / 64), 32, 0, stream>>>(bufC, 32, C2p, H2, out, H2, 32);
}